// MYGRUDecoder_3839700762877
// MI455X (gfx1250) — compile-verified
//
#include <hip/hip_runtime.h>
#include <math.h>

typedef float v2f __attribute__((ext_vector_type(2)));
typedef float v8f __attribute__((ext_vector_type(8)));

// D = A(16x4,f32) * B(4x16,f32) + C(16x16,f32)  -- exact fp32 matrix op
#define WMMA_F32(a, b, c) \
  __builtin_amdgcn_wmma_f32_16x16x4_f32(false, (a), false, (b), (short)0, (c), false, false)

__device__ __forceinline__ float fast_sigmoid(float x) {
  return 1.0f / (1.0f + __expf(-x));
}
__device__ __forceinline__ float fast_tanh(float x) {
  float t = __expf(-2.0f * fabsf(x));       // in (0,1], no overflow
  float r = (1.0f - t) / (1.0f + t);
  return copysignf(r, x);
}

// ---------------------------------------------------------------------------
// Kernel 1: hV[l, b, n] = sum_d H[b, 511-l, d] * V[d, n],  V = [V_r|V_z|V_s]
// Grid: 4096 M-tiles (16 rows of (l,b), fixed l) x 12 N-chunks of 128 cols.
// Block: 256 threads = 8 waves; A tile staged in LDS, each wave one 16x16 tile.
// ---------------------------------------------------------------------------
__global__ __launch_bounds__(256) void hv_gemm(
    const float* __restrict__ H,
    const float* __restrict__ Vr, const float* __restrict__ Vz,
    const float* __restrict__ Vs,
    float* __restrict__ hv)
{
  __shared__ float As[16 * 512];   // 32 KB
  const int tid    = threadIdx.x;
  const int mt     = blockIdx.x / 12;
  const int nchunk = blockIdx.x % 12;
  const int l      = mt >> 3;        // 8 row-tiles per l (128 batch / 16)
  const int bbase  = (mt & 7) << 4;

  // Stage A: As[i][k] = H[bbase+i][511-l][k]  (coalesced along k)
  for (int idx = tid; idx < 16 * 512; idx += 256) {
    int i = idx >> 9, k = idx & 511;
    As[idx] = H[(((size_t)(bbase + i) * 512) + (size_t)(511 - l)) * 512 + k];
  }
  __syncthreads();

  const int wave = tid >> 5, lane = tid & 31;
  const int lm = lane & 15, lh = lane >> 4;
  const int n0 = nchunk * 128 + wave * 16;
  const float* Vp = (n0 < 512) ? Vr : (n0 < 1024 ? Vz : Vs);
  const int c0 = (n0 & 511) + lm;

  v8f acc = {};
  const float* arow = &As[lm * 512 + 2 * lh];
  #pragma unroll 4
  for (int k0 = 0; k0 < 512; k0 += 4) {
    v2f a = *(const v2f*)(arow + k0);             // A[m, k0+2lh], A[m, k0+2lh+1]
    int kb = k0 + 2 * lh;
    v2f b;
    b.x = Vp[(size_t)kb * 512 + c0];
    b.y = Vp[(size_t)(kb + 1) * 512 + c0];
    acc = WMMA_F32(a, b, acc);
  }
  const size_t rowbase = (size_t)l * 128 + bbase;
  #pragma unroll
  for (int j = 0; j < 8; ++j) {
    int m = j + 8 * lh;
    hv[(rowbase + m) * 1536 + n0 + lm] = acc[j];
  }
}

// ---------------------------------------------------------------------------
// Kernel 2: sequential GRU scan. 8 WGs x 1024 threads; WG owns 16 batch rows.
// Wave w owns hidden columns [16w, 16w+16). All matmuls via fp32 WMMA, state
// in LDS, per-step sync with workgroup barriers only. hV tile for the current
// step is prefetched (global_prefetch_b8) at step start so the scattered
// reads at the end of Phase A hit cache.
// ---------------------------------------------------------------------------
__global__ __launch_bounds__(1024) void gru_scan(
    const float* __restrict__ H,  const float* __restrict__ W,
    const float* __restrict__ Bb, const float* __restrict__ U,
    const float* __restrict__ Wx, const float* __restrict__ bx,
    const float* __restrict__ hv, float* __restrict__ out)
{
  __shared__ float sA [16 * 512];  // state s            (32 KB)
  __shared__ float rsA[16 * 512];  // r * s              (32 KB)
  __shared__ float xA [16 * 256];  // x                  (16 KB)

  const int tid   = threadIdx.x;
  const int bbase = blockIdx.x * 16;

  // init: s0 = 0 ; x0 = H[b,511,:256] + H[b,511,256:512]
  for (int idx = tid; idx < 16 * 512; idx += 1024) sA[idx] = 0.0f;
  for (int idx = tid; idx < 16 * 256; idx += 1024) {
    int m = idx >> 8, o = idx & 255;
    const float* h = H + (((size_t)(bbase + m) * 512) + 511) * 512;
    xA[idx] = h[o] + h[256 + o];
  }
  __syncthreads();

  const int wave = tid >> 5, lane = tid & 31;
  const int lm = lane & 15, lh = lane >> 4;
  const int h0 = wave * 16;
  const int c  = h0 + lm;          // this lane's column within [0,512)

  for (int t = 0; t < 512; ++t) {
    const float* hVt = hv + ((size_t)t * 128 + bbase) * 1536;

    // Prefetch this step's 96 KB hV tile: 768 cachelines of 128 B, one per
    // thread. Consumed only after ~192 WMMA k-iterations -> latency hidden.
    if (tid < 768) __builtin_prefetch(hVt + tid * 32, 0, 0);

    // ---- Phase A: pre = x@W + Bb + hV_t ; += s@U  (r, z, s-cand columns) ----
    v8f accR = {}, accZ = {}, accS = {};
    {
      const float* arow = &xA[lm * 256 + 2 * lh];
      #pragma unroll 2
      for (int k0 = 0; k0 < 256; k0 += 4) {            // K over x (256)
        v2f a = *(const v2f*)(arow + k0);
        int kb = k0 + 2 * lh;
        const float* B0 = W + (size_t)kb * 1536 + c;
        const float* B1 = B0 + 1536;
        v2f b;
        b.x = B0[0];    b.y = B1[0];    accR = WMMA_F32(a, b, accR);
        b.x = B0[512];  b.y = B1[512];  accZ = WMMA_F32(a, b, accZ);
        b.x = B0[1024]; b.y = B1[1024]; accS = WMMA_F32(a, b, accS);
      }
      arow = &sA[lm * 512 + 2 * lh];
      #pragma unroll 2
      for (int k0 = 0; k0 < 512; k0 += 4) {            // K over s (512)
        v2f a = *(const v2f*)(arow + k0);
        int kb = k0 + 2 * lh;
        const float* B0 = U + (size_t)kb * 1536 + c;
        const float* B1 = B0 + 1536;
        v2f b;
        b.x = B0[0];    b.y = B1[0];    accR = WMMA_F32(a, b, accR);
        b.x = B0[512];  b.y = B1[512];  accZ = WMMA_F32(a, b, accZ);
        b.x = B0[1024]; b.y = B1[1024]; accS = WMMA_F32(a, b, accS);
      }
    }
    float zreg[8], spreg[8];
    #pragma unroll
    for (int j = 0; j < 8; ++j) {
      int m = j + 8 * lh;
      float pr = accR[j] + Bb[c]        + hVt[(size_t)m * 1536 + c];
      float pz = accZ[j] + Bb[512 + c]  + hVt[(size_t)m * 1536 + 512 + c];
      float ps = accS[j] + Bb[1024 + c] + hVt[(size_t)m * 1536 + 1024 + c];
      float r  = fast_sigmoid(pr);
      zreg[j]  = fast_sigmoid(pz);
      spreg[j] = ps;
      rsA[m * 512 + c] = r * sA[m * 512 + c];
    }
    __syncthreads();

    // ---- Phase B: tmp_s = (r*s) @ U_s ; s = (1-z)*s + z*tanh(ps + tmp_s) ----
    v8f accB = {};
    {
      const float* arow = &rsA[lm * 512 + 2 * lh];
      #pragma unroll 2
      for (int k0 = 0; k0 < 512; k0 += 4) {
        v2f a = *(const v2f*)(arow + k0);
        int kb = k0 + 2 * lh;
        v2f b;
        b.x = U[(size_t)kb * 1536 + 1024 + c];
        b.y = U[(size_t)(kb + 1) * 1536 + 1024 + c];
        accB = WMMA_F32(a, b, accB);
      }
    }
    #pragma unroll
    for (int j = 0; j < 8; ++j) {
      int m = j + 8 * lh;
      float s1   = fast_tanh(spreg[j] + accB[j]);
      float sold = sA[m * 512 + c];
      float z    = zreg[j];
      sA[m * 512 + c] = (1.0f - z) * sold + z * s1;
    }
    __syncthreads();

    // ---- Phase C: x = tanh(s @ Wxo + bx) ; emit output (waves 0..15) ----
    if (wave < 16) {
      const int o = h0 + lm;                   // output col, o0 = wave*16
      v8f accX = {};
      const float* arow = &sA[lm * 512 + 2 * lh];
      #pragma unroll 2
      for (int k0 = 0; k0 < 512; k0 += 4) {
        v2f a = *(const v2f*)(arow + k0);
        int kb = k0 + 2 * lh;
        v2f b;
        b.x = Wx[(size_t)kb * 768 + o];
        b.y = Wx[(size_t)(kb + 1) * 768 + o];
        accX = WMMA_F32(a, b, accX);
      }
      #pragma unroll
      for (int j = 0; j < 8; ++j) {
        int m = j + 8 * lh;
        float xv = fast_tanh(accX[j] + bx[o]);
        xA[m * 256 + o] = xv;
        out[((size_t)(bbase + m) * 512 + (size_t)t) * 256 + o] = xv;
      }
    }
    __syncthreads();
  }
}

// ---------------------------------------------------------------------------
extern "C" void kernel_launch(void* const* d_in, const int* in_sizes, int n_in,
                              void* d_out, int out_size, void* d_ws, size_t ws_size,
                              hipStream_t stream) {
  const float* H  = (const float*)d_in[0];
  const float* W  = (const float*)d_in[1];
  const float* Bb = (const float*)d_in[2];
  const float* U  = (const float*)d_in[3];
  const float* Wx = (const float*)d_in[4];
  const float* bx = (const float*)d_in[5];
  const float* Vr = (const float*)d_in[6];
  const float* Vz = (const float*)d_in[7];
  const float* Vs = (const float*)d_in[8];
  float* out = (float*)d_out;
  float* hv  = (float*)d_ws;   // needs 512*128*1536*4 = 384 MiB scratch

  // Phase 1: fully parallel hV GEMM across the whole GPU
  hv_gemm<<<dim3(4096 * 12), dim3(256), 0, stream>>>(H, Vr, Vz, Vs, hv);
  // Phase 2: 8 independent 16-row GRU scans (batch-decomposed, no global sync)
  gru_scan<<<dim3(8), dim3(1024), 0, stream>>>(H, W, Bb, U, Wx, bx, hv, out);
}